// Net_4793183502401
// MI455X (gfx1250) — compile-verified
//
#include <hip/hip_runtime.h>

typedef __attribute__((ext_vector_type(2))) float v2f;
typedef __attribute__((ext_vector_type(8))) float v8f;

// Hardware FP32 atomic add (global_atomic_add_f32, no return -> STOREcnt only)
__device__ __forceinline__ void atom_add_f32(float* p, float v) {
    __hip_atomic_fetch_add(p, v, __ATOMIC_RELAXED, __HIP_MEMORY_SCOPE_AGENT);
}

__global__ void zero_f32(float* __restrict__ p, int n) {
    int i = blockIdx.x * blockDim.x + threadIdx.x;
    if (i < n) p[i] = 0.0f;
}

// deg[dst] += 1 for every edge
__global__ void degree_kernel(const long long* __restrict__ ei, float* __restrict__ deg, int E) {
    int e = blockIdx.x * blockDim.x + threadIdx.x;
    if (e < E) {
        int dst = (int)ei[(long long)E + e];
        atom_add_f32(&deg[dst], 1.0f);
    }
}

// dinv[i] = rsqrt(deg[i] + 1)   (self-loop; >= 1 so no clamp needed), in place
__global__ void dinv_kernel(float* __restrict__ d, int n) {
    int i = blockIdx.x * blockDim.x + threadIdx.x;
    if (i < n) d[i] = rsqrtf(d[i] + 1.0f);
}

// y1[i,0:16] = dinv[i] * (x[i,0:5] @ W1[5,16])  -- one wave per 16-row tile, WMMA f32 16x16x4.
// dinv is folded into the A rows (scaling A row m scales D row m), so the store
// epilogue is pure stores.
__global__ void transform1_wmma(const float* __restrict__ x, const float* __restrict__ W1,
                                const float* __restrict__ dinv, float* __restrict__ y1, int n) {
    int wave = (blockIdx.x * blockDim.x + threadIdx.x) >> 5;
    int lane = threadIdx.x & 31;
    int ntiles = (n + 15) >> 4;
    if (wave >= ntiles) return;            // wave-uniform: EXEC stays all-ones
    int m0  = wave << 4;
    int rlo = lane & 15;
    int hi  = lane >> 4;
    int row = m0 + rlo;
    int rc  = row < n ? row : n - 1;       // clamp so loads stay unconditional
    int kb  = hi * 2;                      // A/B K stripe: {0,1} lo half, {2,3} hi half
    int col = rlo;
    float di = dinv[rc];                   // scales this lane's A row

    // K-step 0: K = kb, kb+1 (all < 5, valid)
    v2f a0; a0.x = di * x[rc * 5 + kb];   a0.y = di * x[rc * 5 + kb + 1];
    v2f b0; b0.x = W1[kb * 16 + col];     b0.y = W1[(kb + 1) * 16 + col];
    // K-step 1: K = 4+kb, 4+kb+1 ; only K==4 (hi==0, .x) in-range, rest zero-pad
    float xa = di * x[rc * 5 + 4];
    float wa = W1[4 * 16 + col];
    v2f a1; a1.x = (hi == 0) ? xa : 0.0f; a1.y = 0.0f;
    v2f b1; b1.x = (hi == 0) ? wa : 0.0f; b1.y = 0.0f;

    v8f c = {};
    c = __builtin_amdgcn_wmma_f32_16x16x4_f32(false, a0, false, b0, (short)0, c, false, false);
    c = __builtin_amdgcn_wmma_f32_16x16x4_f32(false, a1, false, b1, (short)0, c, false, false);

    // D layout: VGPR r -> M = r + 8*hi, N = col
    if (m0 + 16 <= n) {                    // wave-uniform fast path: full tile, no guards
        #pragma unroll
        for (int r = 0; r < 8; ++r)
            y1[(m0 + r + 8 * hi) * 16 + col] = c[r];
    } else {
        #pragma unroll
        for (int r = 0; r < 8; ++r) {
            int node = m0 + r + 8 * hi;
            if (node < n) y1[node * 16 + col] = c[r];
        }
    }
}

// s1[dst,f] += y1[src,f] ; 16 lanes per edge -> 64B coalesced gather + coalesced atomics
__global__ void scatter1(const long long* __restrict__ ei, const float* __restrict__ y1,
                         float* __restrict__ s1, int E) {
    long long t = (long long)blockIdx.x * blockDim.x + threadIdx.x;
    int e = (int)(t >> 4);
    int f = (int)(t & 15);
    if (e < E) {
        int src = (int)ei[e];
        int dst = (int)ei[(long long)E + e];
        atom_add_f32(&s1[dst * 16 + f], y1[src * 16 + f]);
    }
}

// h[i,k] = relu(dinv[i]*(s1[i,k]+y1[i,k]) + b1[k]);  y2[i,0:4] = dinv[i] * (h[i,:] @ W2[16,4])
// Fused: h built directly in WMMA A layout (lane=row, K striped across half-waves);
// the trailing dinv[i] is folded into the A rows; 4 K-steps of 16x16x4 vs W2 zero-padded
// to 16 output columns.
__global__ void transform2_wmma(const float* __restrict__ s1, const float* __restrict__ y1,
                                const float* __restrict__ W2, const float* __restrict__ b1,
                                const float* __restrict__ dinv, float* __restrict__ y2, int n) {
    int wave = (blockIdx.x * blockDim.x + threadIdx.x) >> 5;
    int lane = threadIdx.x & 31;
    int ntiles = (n + 15) >> 4;
    if (wave >= ntiles) return;            // wave-uniform
    int m0  = wave << 4;
    int rlo = lane & 15;
    int hi  = lane >> 4;
    int row = m0 + rlo;
    int rc  = row < n ? row : n - 1;
    float di = dinv[rc];
    int kb  = hi * 2;
    int col = rlo;
    int c4  = col & 3;                     // clamped W2 column for unconditional load

    v8f c = {};
    #pragma unroll
    for (int s = 0; s < 4; ++s) {
        int k0 = s * 4 + kb;               // even -> 8B-aligned float2 loads
        v2f sv = *(const v2f*)(s1 + rc * 16 + k0);
        v2f yv = *(const v2f*)(y1 + rc * 16 + k0);
        v2f a;
        a.x = di * fmaxf(di * (sv.x + yv.x) + b1[k0],     0.0f);
        a.y = di * fmaxf(di * (sv.y + yv.y) + b1[k0 + 1], 0.0f);
        float w0 = W2[k0 * 4 + c4];
        float w1 = W2[(k0 + 1) * 4 + c4];
        v2f b;
        b.x = (col < 4) ? w0 : 0.0f;
        b.y = (col < 4) ? w1 : 0.0f;
        c = __builtin_amdgcn_wmma_f32_16x16x4_f32(false, a, false, b, (short)0, c, false, false);
    }

    if (m0 + 16 <= n) {                    // wave-uniform fast path
        if (col < 4) {                     // single divergent region with 8 plain stores
            #pragma unroll
            for (int r = 0; r < 8; ++r)
                y2[(m0 + r + 8 * hi) * 4 + col] = c[r];
        }
    } else {
        if (col < 4) {
            #pragma unroll
            for (int r = 0; r < 8; ++r) {
                int node = m0 + r + 8 * hi;
                if (node < n) y2[node * 4 + col] = c[r];
            }
        }
    }
}

// s2[dst,f] += y2[src,f] ; 4 lanes per edge
__global__ void scatter2(const long long* __restrict__ ei, const float* __restrict__ y2,
                         float* __restrict__ s2, int E) {
    long long t = (long long)blockIdx.x * blockDim.x + threadIdx.x;
    int e = (int)(t >> 2);
    int f = (int)(t & 3);
    if (e < E) {
        int src = (int)ei[e];
        int dst = (int)ei[(long long)E + e];
        atom_add_f32(&s2[dst * 4 + f], y2[src * 4 + f]);
    }
}

// out[i,f] = relu(dinv[i]*(s2[i,f]+y2[i,f]) + b2[f])
__global__ void finalize2(const float* __restrict__ s2, const float* __restrict__ y2,
                          const float* __restrict__ b2, const float* __restrict__ dinv,
                          float* __restrict__ out, int n4) {
    int t = blockIdx.x * blockDim.x + threadIdx.x;
    if (t < n4) {
        int i = t >> 2, f = t & 3;
        out[t] = fmaxf(dinv[i] * (s2[t] + y2[t]) + b2[f], 0.0f);
    }
}

static inline int cdiv(long long a, int b) { return (int)((a + b - 1) / b); }

extern "C" void kernel_launch(void* const* d_in, const int* in_sizes, int n_in,
                              void* d_out, int out_size, void* d_ws, size_t ws_size,
                              hipStream_t stream) {
    const float*     x  = (const float*)d_in[0];
    const long long* ei = (const long long*)d_in[1];   // int64 edge_index [2, E]
    const float*     W1 = (const float*)d_in[2];
    const float*     b1 = (const float*)d_in[3];
    const float*     W2 = (const float*)d_in[4];
    const float*     b2 = (const float*)d_in[5];
    float* out = (float*)d_out;

    const int N = in_sizes[0] / 5;    // 250000
    const int E = in_sizes[1] / 2;    // 5000000

    // Workspace layout (floats): 41*N total (~39.1 MiB)
    float* ws   = (float*)d_ws;
    float* dinv = ws;                 // N   (used as deg first, then dinv in place)
    float* y1   = dinv + N;           // 16N
    float* s1   = y1 + 16 * N;        // 16N
    float* y2   = s1 + 16 * N;        // 4N
    float* s2   = y2 + 4 * N;         // 4N

    const int B = 256;

    // 1) zero accumulators
    zero_f32<<<cdiv(N, B),        B, 0, stream>>>(dinv, N);
    zero_f32<<<cdiv(16LL * N, B), B, 0, stream>>>(s1, 16 * N);
    zero_f32<<<cdiv(4LL * N, B),  B, 0, stream>>>(s2, 4 * N);

    // 2) degree, 3) dinv
    degree_kernel<<<cdiv(E, B), B, 0, stream>>>(ei, dinv, E);
    dinv_kernel<<<cdiv(N, B), B, 0, stream>>>(dinv, N);

    // 4) y1 = dinv * (x @ W1)    (WMMA, 1 wave per 16 rows)
    {
        int ntiles = (N + 15) >> 4;
        long long threads = (long long)ntiles * 32;
        transform1_wmma<<<cdiv(threads, 128), 128, 0, stream>>>(x, W1, dinv, y1, N);
    }

    // 5) edge aggregation layer 1
    scatter1<<<cdiv((long long)E * 16, B), B, 0, stream>>>(ei, y1, s1, E);

    // 6) h = relu(dinv*(s1+y1)+b1); y2 = dinv * (h @ W2)   (fused WMMA)
    {
        int ntiles = (N + 15) >> 4;
        long long threads = (long long)ntiles * 32;
        transform2_wmma<<<cdiv(threads, 128), 128, 0, stream>>>(s1, y1, W2, b1, dinv, y2, N);
    }

    // 7) edge aggregation layer 2
    scatter2<<<cdiv((long long)E * 4, B), B, 0, stream>>>(ei, y2, s2, E);

    // 8) out = relu(dinv*(s2+y2)+b2)
    finalize2<<<cdiv(4LL * N, B), B, 0, stream>>>(s2, y2, b2, dinv, out, 4 * N);
}